// SVD_57836029608099
// MI455X (gfx1250) — compile-verified
//
#include <hip/hip_runtime.h>
#include <hip/hip_bf16.h>

// Weighted 2D Procrustes (Kabsch) for B=4096 batches of N=1024 points.
// HBM-bound: ~84MB streamed once (~3.6us at 23.3 TB/s). One wave32 per batch.
// Each lane owns adjacent point pairs -> b128/b64 loads (widest vmem).
// Per-lane f32 partial sums in two banks (even/odd point of the pair),
// reduced across the wave with V_WMMA_F32_16X16X4_F32 (A = partials,
// B = ones -> row sums), then a closed-form 2x2 polar decomposition
// replaces the 3x3 SVD exactly (W's third row/col are identically zero).

typedef float v2f __attribute__((ext_vector_type(2)));
typedef float v8f __attribute__((ext_vector_type(8)));

#define RES  0.2592f
#define CMIN 82.8144f   /* (640/2 - 0.5) * 0.2592 */
#define NPTS 1024
#define NPAIRS (NPTS / 2)

// Reduce 64 f32 partials (2 banks x 32 lanes) to one total, broadcast to all
// lanes. A-matrix 16x4 layout (ISA 7.12.2): lane m%16, vgpr{0,1} hold
// K = 2*(lane/16) + {0,1}. With B == ones(4x16): D[m,n] = P0[m]+P1[m]+
// P0[m+16]+P1[m+16]. Summing the 8 D vgprs gives half-row sums in each lane
// half; one xor-16 shuffle-add completes the exact 32-lane reduction.
__device__ __forceinline__ float wave_reduce_wmma(float p0, float p1) {
    v2f a; a.x = p0; a.y = p1;
    v2f bones; bones.x = 1.0f; bones.y = 1.0f;
    v8f c = {0.f, 0.f, 0.f, 0.f, 0.f, 0.f, 0.f, 0.f};
    v8f d = __builtin_amdgcn_wmma_f32_16x16x4_f32(
        /*neg_a=*/false, a, /*neg_b=*/false, bones,
        /*c_mod=*/(short)0, c, /*reuse_a=*/false, /*reuse_b=*/false);
    float s = d[0] + d[1] + d[2] + d[3] + d[4] + d[5] + d[6] + d[7];
    s += __shfl_xor(s, 16, 32);
    return s;
}

// Accumulate one point into a 9-element moment bank:
// u = [1, trx, try], v = [w, w*srx, w*sry], M += u (outer) v.
#define ACC(S, sx, sy, tx_, ty_, w) do {                                    \
    float srx = CMIN - RES * (sy);                                          \
    float sry = RES * (sx) - CMIN;                                          \
    float trx = CMIN - RES * (ty_);                                         \
    float trY = RES * (tx_) - CMIN;                                         \
    float a1 = (w) * srx, a2 = (w) * sry;                                   \
    S[0] += (w);                                                            \
    S[1] += a1;                                                             \
    S[2] += a2;                                                             \
    S[3] = fmaf((w), trx, S[3]);                                            \
    S[4] = fmaf(trx, a1, S[4]);                                             \
    S[5] = fmaf(trx, a2, S[5]);                                             \
    S[6] = fmaf((w), trY, S[6]);                                            \
    S[7] = fmaf(trY, a1, S[7]);                                             \
    S[8] = fmaf(trY, a2, S[8]);                                             \
} while (0)

__global__ __launch_bounds__(256) void procrustes_svd_kernel(
    const float* __restrict__ kp,   // (B*2, N, 2) pixel coords, src = frames ::2
    const float* __restrict__ tg,   // (B, N, 2)
    const float* __restrict__ wt,   // (B, 1, N)
    float* __restrict__ out,        // [B*9 R^T] ++ [B*3 t]
    int B)
{
    const int lane = threadIdx.x & 31;
    const int wave = threadIdx.x >> 5;
    const int b = blockIdx.x * 8 + wave;
    if (b >= B) return;  // whole-wave uniform (never taken for B=4096)

    // One float4 = one adjacent point pair (x0,y0,x1,y1): b128 loads.
    const float4* __restrict__ sp4 =
        (const float4*)kp + (size_t)(2 * b) * NPAIRS;    // src frame = 2b
    const float4* __restrict__ tp4 =
        (const float4*)tg + (size_t)b * NPAIRS;
    const float2* __restrict__ wp2 =
        (const float2*)wt + (size_t)b * NPAIRS;          // b64 loads

    float A0[9], A1[9];
#pragma unroll
    for (int j = 0; j < 9; ++j) { A0[j] = 0.f; A1[j] = 0.f; }

    // 16 iterations; per lane: 1x b128 src + 1x b128 tgt + 1x b64 weights.
#pragma unroll 4
    for (int p = lane; p < NPAIRS; p += 32) {
        float4 s4 = sp4[p];
        float4 t4 = tp4[p];
        float2 w2 = wp2[p];
        ACC(A0, s4.x, s4.y, t4.x, t4.y, w2.x);   // even point -> bank 0
        ACC(A1, s4.z, s4.w, t4.z, t4.w, w2.y);   // odd point  -> bank 1
    }

    // Exact cross-lane reduction on the matrix unit (EXEC all-ones here).
    float T[9];
#pragma unroll
    for (int j = 0; j < 9; ++j) T[j] = wave_reduce_wmma(A0[j], A1[j]);

    if (lane == 0) {
        const float inv = 1.0f / T[0];
        const float scx = T[1] * inv, scy = T[2] * inv;   // src centroid
        const float tcx = T[3] * inv, tcy = T[6] * inv;   // tgt centroid

        // W2[i][j] = E[w * tgt_i * src_j]/w - tc_i*sc_j  (2x2 block of W)
        const float a  = T[4] * inv - tcx * scx;
        const float bb = T[5] * inv - tcx * scy;
        const float cc = T[7] * inv - tcy * scx;
        const float dd = T[8] * inv - tcy * scy;
        const float det = a * dd - bb * cc;

        // Orthogonal polar factor of W2 == U2*V2^T from the SVD; R[2][2] =
        // det(U2)det(V2) = sign(det W2). Identical to U diag(1,1,detUV) V^T.
        float r00, r01, r10, r11, r22;
        if (det >= 0.0f) {
            float p = a + dd, q = cc - bb;
            float rl = fmaxf(sqrtf(p * p + q * q), 1e-30f);
            p /= rl; q /= rl;
            r00 = p;  r01 = -q;
            r10 = q;  r11 = p;
            r22 = 1.0f;
        } else {
            float p = a - dd, q = bb + cc;
            float rl = fmaxf(sqrtf(p * p + q * q), 1e-30f);
            p /= rl; q /= rl;
            r00 = p;  r01 = q;
            r10 = q;  r11 = -p;
            r22 = -1.0f;
        }

        // t1 = src_c - R^T * tgt_c ; t = -R * t1 (z components are exactly 0)
        const float t1x = scx - (r00 * tcx + r10 * tcy);
        const float t1y = scy - (r01 * tcx + r11 * tcy);
        const float tx = -(r00 * t1x + r01 * t1y);
        const float ty = -(r10 * t1x + r11 * t1y);

        float* __restrict__ R = out + (size_t)b * 9;   // stores R^T row-major
        R[0] = r00; R[1] = r10; R[2] = 0.0f;
        R[3] = r01; R[4] = r11; R[5] = 0.0f;
        R[6] = 0.0f; R[7] = 0.0f; R[8] = r22;

        float* __restrict__ tvec = out + (size_t)B * 9 + (size_t)b * 3;
        tvec[0] = tx; tvec[1] = ty; tvec[2] = 0.0f;
    }
}

extern "C" void kernel_launch(void* const* d_in, const int* in_sizes, int n_in,
                              void* d_out, int out_size, void* d_ws, size_t ws_size,
                              hipStream_t stream) {
    const float* kp = (const float*)d_in[0];  // keypoint_coords (B*2, N, 2)
    const float* tg = (const float*)d_in[1];  // tgt_coords      (B,   N, 2)
    const float* wt = (const float*)d_in[2];  // weights         (B, 1, N)
    float* out = (float*)d_out;

    const int B = in_sizes[2] / NPTS;         // weights holds B*N floats
    const int blocks = (B + 7) / 8;           // 8 waves (batches) per block
    procrustes_svd_kernel<<<blocks, 256, 0, stream>>>(kp, tg, wt, out, B);
}